// BipartiteGraphLayer_4415226380865
// MI455X (gfx1250) — compile-verified
//
#include <hip/hip_runtime.h>

typedef float v2f __attribute__((ext_vector_type(2)));
typedef float v8f __attribute__((ext_vector_type(8)));

#define D 256
#define LN_EPS 1e-5f

// ---------------------------------------------------------------- zero fill
__global__ void zero_ws_kernel(float* __restrict__ ws, long n) {
  long i = (long)blockIdx.x * blockDim.x + threadIdx.x;
  long stride = (long)gridDim.x * blockDim.x;
  for (; i < n; i += stride) ws[i] = 0.0f;
}

// ------------------------------------------------ 256x256 transpose (W -> Wt)
// Wt[n][k] = W[k][n]; makes WMMA B-fragment K-pairs contiguous (b64 loads).
__global__ __launch_bounds__(256) void transpose_w_kernel(
    const float* __restrict__ W, float* __restrict__ Wt) {
  __shared__ float tile[32][33];
  int bx = blockIdx.x & 7;        // col block of W
  int by = blockIdx.x >> 3;       // row block of W
  int tx = threadIdx.x & 31;
  int ty = threadIdx.x >> 5;      // 0..7
#pragma unroll
  for (int r = 0; r < 32; r += 8)
    tile[ty + r][tx] = W[(by * 32 + ty + r) * D + bx * 32 + tx];
  __syncthreads();
#pragma unroll
  for (int r = 0; r < 32; r += 8)
    Wt[(bx * 32 + ty + r) * D + by * 32 + tx] = tile[tx][ty + r];
}

// --------------------------------------------------- edge scatter (segment sum)
// 4 edges per 256-thread block; 64 lanes per edge; float4 gather from source
// row + 4 global_atomic_add_f32 into the target row; lane 0 bumps degree.
__global__ void edge_scatter_kernel(const float* __restrict__ src_feat,
                                    const int* __restrict__ edges, int E,
                                    float* __restrict__ agg,
                                    float* __restrict__ deg) {
  int slot = threadIdx.x >> 6;   // 0..3
  int t    = threadIdx.x & 63;   // 0..63 -> 4 floats each
  int e = blockIdx.x * 4 + slot;
  if (e >= E) return;
  int s = edges[e];        // source node index (row 0)
  int d = edges[E + e];    // target node index (row 1)
  float4 v = ((const float4*)(src_feat + (long)s * D))[t];
  float* drow = agg + (long)d * D + t * 4;
  unsafeAtomicAdd(drow + 0, v.x);
  unsafeAtomicAdd(drow + 1, v.y);
  unsafeAtomicAdd(drow + 2, v.z);
  unsafeAtomicAdd(drow + 3, v.w);
  if (t == 0) unsafeAtomicAdd(deg + d, 1.0f);
}

// ------------------------------------------- X = feat + agg / max(deg,1), in place
__global__ void combine_kernel(const float* __restrict__ feat,
                               float* __restrict__ agg,
                               const float* __restrict__ deg, long n4) {
  long i = (long)blockIdx.x * blockDim.x + threadIdx.x;
  long stride = (long)gridDim.x * blockDim.x;
  for (; i < n4; i += stride) {
    long row = i >> 6;  // D/4 = 64 float4 per row
    float dinv = 1.0f / fmaxf(deg[row], 1.0f);
    float4 f = ((const float4*)feat)[i];
    float4 a = ((float4*)agg)[i];
    a.x = f.x + a.x * dinv;
    a.y = f.y + a.y * dinv;
    a.z = f.z + a.z * dinv;
    a.w = f.w + a.w * dinv;
    ((float4*)agg)[i] = a;
  }
}

// --------------------------- out = LayerNorm(ReLU(X @ W + b)) with fp32 WMMA
// Block: 128 threads = 4 waves, computing rows [m0, m0+32) x all 256 cols.
// Wave w owns cols [w*64, w*64+64): 2 M-tiles x 4 N-tiles = 8 accumulators.
// Per K-step(4): 2 A b64 loads + 4 B b64 loads -> 8 WMMAs (B reused across M).
#define WMMA4(A, B, C) \
  C = __builtin_amdgcn_wmma_f32_16x16x4_f32(false, A, false, B, (short)0, C, false, false)

__global__ __launch_bounds__(128) void gemm_bias_relu_ln_kernel(
    const float* __restrict__ X, const float* __restrict__ Wt,
    const float* __restrict__ bias, const float* __restrict__ gamma,
    const float* __restrict__ beta, float* __restrict__ out, int M) {
  __shared__ float sTile[32][D + 8];
  __shared__ float sSum[32][4];
  __shared__ float sSq[32][4];
  __shared__ float sMu[32];
  __shared__ float sRstd[32];

  const int tid  = threadIdx.x;
  const int wave = tid >> 5;
  const int lane = tid & 31;
  const int h    = lane >> 4;   // half-wave: K-pair select / C row offset
  const int ln   = lane & 15;   // M within tile (A), N within tile (B/C)
  const long m0  = (long)blockIdx.x * 32;

  v8f c0 = {}, c1 = {}, c2 = {}, c3 = {};   // rows m0..m0+15
  v8f c4 = {}, c5 = {}, c6 = {}, c7 = {};   // rows m0+16..m0+31
  const int nBase = wave * 64 + ln;
  const float* xrow0 = X + (m0 + ln) * D;
  const float* xrow1 = X + (m0 + 16 + ln) * D;
  const float* wrow  = Wt + (long)nBase * D;   // Wt[n][k]

  for (int k0 = 0; k0 < D; k0 += 4) {
    const int kk = k0 + 2 * h;
    v2f a0 = *(const v2f*)(xrow0 + kk);
    v2f a1 = *(const v2f*)(xrow1 + kk);
    v2f b0 = *(const v2f*)(wrow + kk);
    v2f b1 = *(const v2f*)(wrow + 16 * D + kk);
    v2f b2 = *(const v2f*)(wrow + 32 * D + kk);
    v2f b3 = *(const v2f*)(wrow + 48 * D + kk);
    WMMA4(a0, b0, c0);
    WMMA4(a0, b1, c1);
    WMMA4(a0, b2, c2);
    WMMA4(a0, b3, c3);
    WMMA4(a1, b0, c4);
    WMMA4(a1, b1, c5);
    WMMA4(a1, b2, c6);
    WMMA4(a1, b3, c7);
  }

  // bias + ReLU, stage C tiles to LDS. C layout: VGPR r -> row r + 8*h.
  {
    auto stage = [&](v8f acc, int toff, int roff) {
      int n = nBase + toff;
      float bn = bias[n];
#pragma unroll
      for (int r = 0; r < 8; ++r) {
        float vv = acc[r] + bn;
        sTile[r + 8 * h + roff][n] = fmaxf(vv, 0.0f);
      }
    };
    stage(c0, 0, 0);
    stage(c1, 16, 0);
    stage(c2, 32, 0);
    stage(c3, 48, 0);
    stage(c4, 0, 16);
    stage(c5, 16, 16);
    stage(c6, 32, 16);
    stage(c7, 48, 16);
  }
  __syncthreads();

  // per-row mean / var: 4 threads per row, 64 cols each
  {
    int row = tid >> 2;
    int j   = tid & 3;
    float s = 0.0f, q = 0.0f;
#pragma unroll 8
    for (int c = j * 64; c < j * 64 + 64; ++c) {
      float v = sTile[row][c];
      s += v;
      q += v * v;
    }
    sSum[row][j] = s;
    sSq[row][j]  = q;
  }
  __syncthreads();
  {
    int row = tid >> 2;
    int j   = tid & 3;
    if (j == 0) {
      float S = 0.0f, Q = 0.0f;
#pragma unroll
      for (int jj = 0; jj < 4; ++jj) { S += sSum[row][jj]; Q += sSq[row][jj]; }
      float mu  = S * (1.0f / D);
      float var = Q * (1.0f / D) - mu * mu;
      sMu[row]   = mu;
      sRstd[row] = rsqrtf(var + LN_EPS);
    }
  }
  __syncthreads();

  // normalize + affine, coalesced write-out (guard padded rows)
  for (int idx = tid; idx < 32 * D; idx += 128) {
    int row = idx >> 8;
    int c   = idx & (D - 1);
    if (m0 + row < M) {
      float v = (sTile[row][c] - sMu[row]) * sRstd[row] * gamma[c] + beta[c];
      out[(m0 + row) * D + c] = v;
    }
  }
}

extern "C" void kernel_launch(void* const* d_in, const int* in_sizes, int n_in,
                              void* d_out, int out_size, void* d_ws, size_t ws_size,
                              hipStream_t stream) {
  const float* user_feat = (const float*)d_in[0];
  const float* item_feat = (const float*)d_in[1];
  const int*   ui_edges  = (const int*)d_in[2];
  const int*   iu_edges  = (const int*)d_in[3];
  const float* Wu    = (const float*)d_in[4];
  const float* bu    = (const float*)d_in[5];
  const float* Wi    = (const float*)d_in[6];
  const float* bi    = (const float*)d_in[7];
  const float* gamma = (const float*)d_in[8];
  const float* beta  = (const float*)d_in[9];

  const long NU = in_sizes[0] / D;
  const long NI = in_sizes[1] / D;
  const int  E  = in_sizes[2] / 2;
  const long NUp = (NU + 31) & ~31L;   // pad to 32 rows for the GEMM tiling
  const long NIp = (NI + 31) & ~31L;

  float* ws    = (float*)d_ws;
  float* agg_u = ws;                    // NUp*D  (becomes X_u; padding stays 0)
  float* agg_i = agg_u + NUp * D;       // NIp*D  (becomes X_i)
  float* deg_u = agg_i + NIp * D;       // NUp
  float* deg_i = deg_u + NUp;           // NIp
  float* wt_u  = deg_i + NIp;           // D*D transposed Wu
  float* wt_i  = wt_u + (long)D * D;    // D*D transposed Wi

  float* out_u = (float*)d_out;
  float* out_i = out_u + NU * D;

  long zn = NUp * D + NIp * D + NUp + NIp;
  zero_ws_kernel<<<2048, 256, 0, stream>>>(ws, zn);

  transpose_w_kernel<<<64, 256, 0, stream>>>(Wu, wt_u);
  transpose_w_kernel<<<64, 256, 0, stream>>>(Wi, wt_i);

  int sblocks = (E + 3) / 4;
  edge_scatter_kernel<<<sblocks, 256, 0, stream>>>(user_feat, ui_edges, E, agg_i, deg_i);
  edge_scatter_kernel<<<sblocks, 256, 0, stream>>>(item_feat, iu_edges, E, agg_u, deg_u);

  combine_kernel<<<2048, 256, 0, stream>>>(user_feat, agg_u, deg_u, NU * D / 4);
  combine_kernel<<<2048, 256, 0, stream>>>(item_feat, agg_i, deg_i, NI * D / 4);

  int gblocks_u = (int)((NU + 31) / 32);
  int gblocks_i = (int)((NI + 31) / 32);
  gemm_bias_relu_ln_kernel<<<gblocks_u, 128, 0, stream>>>(agg_u, wt_u, bu, gamma, beta, out_u, (int)NU);
  gemm_bias_relu_ln_kernel<<<gblocks_i, 128, 0, stream>>>(agg_i, wt_i, bi, gamma, beta, out_i, (int)NI);
}